// TokenChoiceTopKRouter_32993938768150
// MI455X (gfx1250) — compile-verified
//
#include <hip/hip_runtime.h>
#include <hip/hip_bf16.h>
#include <math.h>

typedef __attribute__((ext_vector_type(2))) float v2f;
typedef __attribute__((ext_vector_type(4))) float v4f;
typedef __attribute__((ext_vector_type(8))) float v8f;

#define N_TOKENS 16384
#define HIDDEN   2048
#define EXPERTS  8
#define EPS      1e-20f

// output layout (flat, return order)
#define OFF_TOP   0                        // 16384*2
#define OFF_SCORE (N_TOKENS * 2)           // 16384*8
#define OFF_IDX   (OFF_SCORE + N_TOKENS*8) // 16384*2
#define OFF_CNT   (OFF_IDX + N_TOKENS*2)   // 8
#define OFF_ENT   (OFF_CNT + 8)            // 1

__global__ void router_init_kernel(float* __restrict__ out) {
    int t = threadIdx.x;
    if (t < 9) out[OFF_CNT + t] = 0.0f;    // 8 counts + 1 entropy
}

// grid = 128 blocks * 256 threads; each wave32 handles one 16-token tile.
__global__ void __launch_bounds__(256, 2)
router_kernel(const float* __restrict__ x,
              const float* __restrict__ W,
              const float* __restrict__ bias,
              float* __restrict__ out) {
    __shared__ float s_ent;
    __shared__ int   s_hist[EXPERTS];

    const int tid  = threadIdx.x;
    const int lane = tid & 31;
    const int wave = tid >> 5;

    if (tid == 0) s_ent = 0.0f;
    if (tid < EXPERTS) s_hist[tid] = 0;
    __syncthreads();

    // ---- V_WMMA_F32_16X16X4_F32 fragments ----
    // Canonical layout: A(16x4) / B(4x16): half0 lanes supply K0,K1 ; half1
    // lanes supply K2,K3 (one VGPR each). Since A and B fragments here use the
    // *same* lane/element -> K assignment, any consistent K permutation is a
    // valid partial sum. Each lane loads a float4 of W and of x at
    // K = kc + 4*(lane>>4); elements {0,1} feed WMMA #0, {2,3} feed WMMA #1,
    // covering K = kc..kc+7 exactly once with two b128 loads per two WMMAs.
    // D(16x16): VGPR r, lanes 0-15 -> M=r, N=lane  => lane L<16 ends up with
    // all 8 expert logits of token L in c[0..7].
    const int tile   = blockIdx.x * 8 + wave;        // 1024 tiles of 16 tokens
    const int nt     = lane & 15;                    // token-in-tile / expert id
    const int koff   = (lane >> 4) * 4;
    const int token  = tile * 16 + nt;
    const int expert = nt & 7;                       // D rows 8-15 are discarded

    const float* xrow = x + (size_t)token  * HIDDEN + koff;
    const float* wrow = W + (size_t)expert * HIDDEN + koff;

    v8f c = {};
    #pragma unroll 4
    for (int kc = 0; kc < HIDDEN; kc += 8) {
        v4f wv = *(const v4f*)(wrow + kc);                            // WGP$/L2-hot
        v4f xv = __builtin_nontemporal_load((const v4f*)(xrow + kc)); // HBM stream
        v2f a0 = {wv.x, wv.y}, b0 = {xv.x, xv.y};
        v2f a1 = {wv.z, wv.w}, b1 = {xv.z, xv.w};
        c = __builtin_amdgcn_wmma_f32_16x16x4_f32(
                false, a0, false, b0, (short)0, c, false, false);
        c = __builtin_amdgcn_wmma_f32_16x16x4_f32(
                false, a1, false, b1, (short)0, c, false, false);
    }

    // ---- lane-local router epilogue (lanes 0..15 own one token each) ----
    if (lane < 16) {
        float sc[EXPERTS], bi[EXPERTS];
        #pragma unroll
        for (int e = 0; e < EXPERTS; ++e) {
            float s = __builtin_amdgcn_rcpf(1.0f + __expf(-c[e]));   // sigmoid
            sc[e] = s;
            bi[e] = s + bias[e];
        }
        v4f sv0 = {sc[0], sc[1], sc[2], sc[3]};
        v4f sv1 = {sc[4], sc[5], sc[6], sc[7]};
        *(v4f*)(out + OFF_SCORE + (size_t)token * EXPERTS)     = sv0;
        *(v4f*)(out + OFF_SCORE + (size_t)token * EXPERTS + 4) = sv1;

        // top-2 over biased scores, ties -> lower index (jax top_k order)
        int i0 = 0; float b0 = bi[0];
        #pragma unroll
        for (int e = 1; e < EXPERTS; ++e)
            if (bi[e] > b0) { b0 = bi[e]; i0 = e; }
        int i1 = -1; float b1 = -INFINITY;
        #pragma unroll
        for (int e = 0; e < EXPERTS; ++e)
            if (e != i0 && bi[e] > b1) { b1 = bi[e]; i1 = e; }

        float s0 = sc[i0], s1 = sc[i1];
        float inv = __builtin_amdgcn_rcpf(s0 + s1 + EPS);
        float t0 = s0 * inv, t1 = s1 * inv;

        v2f tv = {t0, t1};
        v2f iv = {(float)i0, (float)i1};
        *(v2f*)(out + OFF_TOP + (size_t)token * 2) = tv;
        *(v2f*)(out + OFF_IDX + (size_t)token * 2) = iv;

        float ent = -(t0 * __logf(t0) + t1 * __logf(t1));
        atomicAdd(&s_ent, ent);                      // ds_add_f32
        atomicAdd(&s_hist[i0], 1);                   // ds_add_u32
        atomicAdd(&s_hist[i1], 1);
    }
    __syncthreads();

    if (tid == 0)
        atomicAdd(&out[OFF_ENT], s_ent * (1.0f / (float)N_TOKENS));
    if (tid < EXPERTS)
        atomicAdd(&out[OFF_CNT + tid], (float)s_hist[tid]);
}

extern "C" void kernel_launch(void* const* d_in, const int* in_sizes, int n_in,
                              void* d_out, int out_size, void* d_ws, size_t ws_size,
                              hipStream_t stream) {
    const float* x    = (const float*)d_in[0];
    const float* W    = (const float*)d_in[1];
    const float* bias = (const float*)d_in[2];
    float* out = (float*)d_out;

    router_init_kernel<<<1, 32, 0, stream>>>(out);
    router_kernel<<<N_TOKENS / 128, 256, 0, stream>>>(x, W, bias, out);
}